// Traversability_24876450579293
// MI455X (gfx1250) — compile-verified
//
#include <hip/hip_runtime.h>
#include <hip/hip_bf16.h>
#include <stdint.h>

// Chamfer / clamped Euclidean distance transform, radius 21.
//   out[b,y,x] = sqrt( min(big, min_{|dx|,|dy|<=21, obstacle} dx^2+dy^2) )
// Horizontal pass degenerates to nearest-obstacle-in-row (binary input);
// vertical pass is a 43-tap min-plus parabola. Fully fused in one kernel
// with an LDS halo tile filled by gfx1250 async global->LDS copies.

#define RTRAV 21
#define TX 64
#define TY 32
#define PITCH (TX + 2 * RTRAV)  // 106
#define ROWS  (TY + 2 * RTRAV)  // 74
#define NTHREADS 256

#if defined(__HIP_DEVICE_COMPILE__) && defined(__gfx1250__) && \
    __has_builtin(__builtin_amdgcn_global_load_async_to_lds_b32)
#define USE_ASYNC_LDS 1
#else
#define USE_ASYNC_LDS 0
#endif

#if USE_ASYNC_LDS
typedef __attribute__((address_space(1))) int gas_int;
typedef __attribute__((address_space(3))) int las_int;
#endif

__global__ __launch_bounds__(NTHREADS)
void Traversability_24876450579293_kernel(const int* __restrict__ obs_g,
                                          const float* __restrict__ max_sq,
                                          float* __restrict__ out,
                                          int H, int W) {
  __shared__ int   obs[ROWS * PITCH];   // 74*106*4 = 31376 B
  __shared__ float hsq[ROWS * TX];      // 74*64*4  = 18944 B

  const int tid = threadIdx.x;
  const int x0  = blockIdx.x * TX;
  const int y0  = blockIdx.y * TY;
  const long long base = (long long)blockIdx.z * H * W;
  const float bigv = max_sq[0];  // 441.0f

  // ---- Stage A: halo tile (async global -> LDS where available) ----
  for (int i = tid; i < ROWS * PITCH; i += NTHREADS) {
    const int r  = i / PITCH;
    const int c  = i - r * PITCH;
    const int gy = y0 + r - RTRAV;
    const int gx = x0 + c - RTRAV;
    if (gy >= 0 && gy < H && gx >= 0 && gx < W) {
      const long long idx = base + (long long)gy * W + gx;
#if USE_ASYNC_LDS
      __builtin_amdgcn_global_load_async_to_lds_b32(
          (gas_int*)(unsigned long long)(uintptr_t)(obs_g + idx),
          (las_int*)(&obs[i]), /*offset=*/0, /*cpol=*/0);
#else
      obs[i] = obs_g[idx];
#endif
    } else {
      obs[i] = 0;  // outside grid: no obstacle (equivalent to big-padding)
    }
  }
#if USE_ASYNC_LDS
#if __has_builtin(__builtin_amdgcn_s_wait_asynccnt)
  __builtin_amdgcn_s_wait_asynccnt(0);
#else
  asm volatile("s_wait_asynccnt 0" ::: "memory");
#endif
#endif
  __syncthreads();

  // ---- Stage B: horizontal squared distance to nearest obstacle (<=21) ----
  for (int i = tid; i < ROWS * TX; i += NTHREADS) {
    const int r = i >> 6;          // / TX
    const int c = i & (TX - 1);
    const int* row = &obs[r * PITCH + c + RTRAV];
    float best = row[0] ? 0.0f : bigv;
#pragma unroll
    for (int d = 1; d <= RTRAV; ++d) {
      const int hit = row[-d] | row[d];
      if (hit) best = fminf(best, (float)(d * d));
    }
    hsq[i] = best;
  }
  __syncthreads();

  // ---- Stage C: vertical 43-tap min-plus parabola + sqrt ----
  const int tx  = tid & (TX - 1);
  const int ty0 = tid >> 6;        // 0..3
  for (int ty = ty0; ty < TY; ty += NTHREADS / TX) {
    float best = bigv;
#pragma unroll
    for (int k = 0; k <= 2 * RTRAV; ++k) {
      const float dsq = (float)((k - RTRAV) * (k - RTRAV));
      best = fminf(best, hsq[(ty + k) * TX + tx] + dsq);
    }
    const long long o = base + (long long)(y0 + ty) * W + (x0 + tx);
    out[o] = __builtin_sqrtf(best);
  }
}

extern "C" void kernel_launch(void* const* d_in, const int* in_sizes, int n_in,
                              void* d_out, int out_size, void* d_ws, size_t ws_size,
                              hipStream_t stream) {
  (void)n_in; (void)out_size; (void)d_ws; (void)ws_size;
  const int* obs      = (const int*)d_in[0];     // bool grid as int32
  const float* max_sq = (const float*)d_in[1];   // {441.0f}
  float* out          = (float*)d_out;

  const int H = 2048, W = 2048;
  int B = in_sizes[0] / (H * W);
  if (B < 1) B = 1;

  dim3 grid(W / TX, H / TY, B);
  Traversability_24876450579293_kernel<<<grid, NTHREADS, 0, stream>>>(
      obs, max_sq, out, H, W);
}